// VisualPromptEncoder_49074296324730
// MI455X (gfx1250) — compile-verified
//
#include <hip/hip_runtime.h>
#include <cmath>
#include <stdint.h>

#define NUM_CLASSES 80
#define HB 160
#define WB 160
#define CCH 256
#define NBOX 100
#define LSTRIDE 164   // padded LDS row stride (floats): rows stay 16B-aligned, banks well spread

typedef float v2f __attribute__((ext_vector_type(2)));
typedef float v8f __attribute__((ext_vector_type(8)));

// Toggle for the CDNA5 async global->LDS staging path (ASYNCcnt / TDM-style copy).
#define USE_ASYNC_LDS 1

__device__ __forceinline__ void async_copy16_to_lds(const float* g, float* l) {
#if USE_ASYNC_LDS
  // LDS byte address = low 32 bits of the generic pointer to __shared__ memory
  // (addrspacecast generic->LDS on AMDGPU truncates; HW adds LDS_BASE).
  const uint32_t lofs = (uint32_t)(uintptr_t)l;
  asm volatile("global_load_async_to_lds_b128 %0, %1, off"
               :
               : "v"(lofs), "v"(g)
               : "memory");
#else
  *(float4*)l = *(const float4*)g;
#endif
}

__device__ __forceinline__ void wait_async_zero() {
#if __has_builtin(__builtin_amdgcn_s_wait_asynccnt)
  __builtin_amdgcn_s_wait_asynccnt(0);
#else
  asm volatile("s_wait_asynccnt 0" ::: "memory");
#endif
}

__global__ __launch_bounds__(256)
void vpe_sat_wmma_kernel(const float* __restrict__ feat,
                         const float* __restrict__ boxes,
                         const int*   __restrict__ gtc,
                         const int*   __restrict__ negy,
                         const int*   __restrict__ negx,
                         const int*   __restrict__ imh,
                         const int*   __restrict__ imw,
                         float*       __restrict__ out)
{
  __shared__ float Sm[HB * LSTRIDE];   // 160x160 plane, padded rows -> SAT in place
  __shared__ float pBox[NBOX];
  __shared__ float vBox[NBOX];
  __shared__ int   cBox[NBOX];

  const int tid  = threadIdx.x;
  const int lane = tid & 31;
  const int wave = tid >> 5;           // 8 waves of 32 (wave32)
  const int b = blockIdx.x >> 8;       // blockIdx.x = b*256 + c
  const int c = blockIdx.x & 255;
  const float* __restrict__ plane = feat + (size_t)(b * CCH + c) * (HB * WB);

  // ---------------- Phase 0: stage plane into LDS (async b128 copies) ----------------
  for (int i = tid; i < (HB * WB) / 4; i += 256) {
    const int row = i / (WB / 4);
    const int q4  = i - row * (WB / 4);
    async_copy16_to_lds(plane + row * WB + q4 * 4, &Sm[row * LSTRIDE + q4 * 4]);
  }
#if USE_ASYNC_LDS
  wait_async_zero();
#endif
  __syncthreads();

  // Fragment lane mapping (V_WMMA_F32_16X16X4_F32, wave32):
  //   A[m][k]:  m = lane&15, k = v + 2*(lane>>4)            (2 VGPRs)
  //   B[k][n]:  n = lane&15, k = v + 2*(lane>>4)            (2 VGPRs)
  //   C/D[m][n]: n = lane&15, m = v + 8*(lane>>4)           (8 VGPRs)
  const int m  = lane & 15;
  const int hi = lane >> 4;

  // Inclusive-triangular fragment per K-chunk q (element k = 4q + 2*hi + {0,1}):
  // as B in the row scan: Tri[k][n] = (k <= n); as A in the col scan: Tri[m][k] = (k <= m).
  v2f tri[4];
#pragma unroll
  for (int q = 0; q < 4; ++q) {
    const int kb = 4 * q + 2 * hi;
    tri[q].x = (kb     <= m) ? 1.0f : 0.0f;
    tri[q].y = (kb + 1 <= m) ? 1.0f : 0.0f;
  }
  const v2f ones = {1.0f, 1.0f};

  // ---------------- Phase 1: row-wise inclusive scan  S = F * U  (one wave per 16-row strip) ----------------
  for (int s = wave; s < 10; s += 8) {
    const int rowA = 16 * s + m;                   // A-fragment row owned by this lane
    v2f run[4] = {{0.f,0.f},{0.f,0.f},{0.f,0.f},{0.f,0.f}};   // running row-sum fragment R
    for (int J = 0; J < 10; ++J) {
      v2f aq[4];
#pragma unroll
      for (int q = 0; q < 4; ++q) {
        const int col = 16 * J + 4 * q + 2 * hi;
        aq[q].x = Sm[rowA * LSTRIDE + col];
        aq[q].y = Sm[rowA * LSTRIDE + col + 1];
      }
      v8f acc = {};
#pragma unroll
      for (int q = 0; q < 4; ++q)   // F(J) * Tri   (within-tile prefix)
        acc = __builtin_amdgcn_wmma_f32_16x16x4_f32(false, aq[q], false, tri[q],
                                                    (short)0, acc, false, false);
#pragma unroll
      for (int q = 0; q < 4; ++q)   // R * Ones     (carry-in from previous tiles)
        acc = __builtin_amdgcn_wmma_f32_16x16x4_f32(false, run[q], false, ones,
                                                    (short)0, acc, false, false);
#pragma unroll
      for (int v = 0; v < 8; ++v)   // store D in place over tile (s, J)
        Sm[(16 * s + v + 8 * hi) * LSTRIDE + 16 * J + m] = acc[v];
#pragma unroll
      for (int q = 0; q < 4; ++q) { run[q].x += aq[q].x; run[q].y += aq[q].y; }
    }
  }
  __syncthreads();

  // ---------------- Phase 2: column-wise inclusive scan  S2 = L * S  (one wave per 16-col strip) ----------------
  for (int xb = wave; xb < 10; xb += 8) {
    const int colB = 16 * xb + m;                  // B-fragment column owned by this lane
    v2f run[4] = {{0.f,0.f},{0.f,0.f},{0.f,0.f},{0.f,0.f}};   // running column-sum fragment
    for (int Jy = 0; Jy < 10; ++Jy) {
      v2f bq[4];
#pragma unroll
      for (int q = 0; q < 4; ++q) {
        const int row = 16 * Jy + 4 * q + 2 * hi;
        bq[q].x = Sm[row * LSTRIDE + colB];
        bq[q].y = Sm[(row + 1) * LSTRIDE + colB];
      }
      v8f acc = {};
#pragma unroll
      for (int q = 0; q < 4; ++q)   // Tri * S(Jy)
        acc = __builtin_amdgcn_wmma_f32_16x16x4_f32(false, tri[q], false, bq[q],
                                                    (short)0, acc, false, false);
#pragma unroll
      for (int q = 0; q < 4; ++q)   // Ones * R
        acc = __builtin_amdgcn_wmma_f32_16x16x4_f32(false, ones, false, run[q],
                                                    (short)0, acc, false, false);
#pragma unroll
      for (int v = 0; v < 8; ++v)
        Sm[(16 * Jy + v + 8 * hi) * LSTRIDE + colB] = acc[v];
#pragma unroll
      for (int q = 0; q < 4; ++q) { run[q].x += bq[q].x; run[q].y += bq[q].y; }
    }
  }
  __syncthreads();

  // ---------------- Phase 3: per-box SAT corner gather -> pooled value ----------------
  if (tid < NBOX) {
    const float sw = (float)WB / (float)imw[0];
    const float sh = (float)HB / (float)imh[0];
    const float* bp = boxes + (size_t)(b * NBOX + tid) * 4;
    int x1 = (int)floorf(bp[0] * sw); x1 = x1 < 0 ? 0 : (x1 > WB ? WB : x1);
    int y1 = (int)floorf(bp[1] * sh); y1 = y1 < 0 ? 0 : (y1 > HB ? HB : y1);
    int x2 = (int)floorf(bp[2] * sw); x2 = x2 < 0 ? 0 : (x2 > WB ? WB : x2);
    int y2 = (int)floorf(bp[3] * sh); y2 = y2 < 0 ? 0 : (y2 > HB ? HB : y2);
    const bool valid = (x2 > x1) && (y2 > y1);
    auto satv = [&](int y, int x) -> float {
      return (y == 0 || x == 0) ? 0.0f : Sm[(y - 1) * LSTRIDE + (x - 1)];
    };
    const float bs = satv(y2, x2) - satv(y1, x2) - satv(y2, x1) + satv(y1, x1);
    int area = (x2 - x1) * (y2 - y1); if (area < 1) area = 1;
    pBox[tid] = valid ? (bs / (float)area) : 0.0f;
    vBox[tid] = valid ? 1.0f : 0.0f;
    cBox[tid] = gtc[b * NBOX + tid];
  }
  __syncthreads();

  // ---------------- Phase 4: deterministic per-class mean (+ negative-pixel fallback) ----------------
  if (tid < NUM_CLASSES) {
    float ws = 0.0f, cnt = 0.0f;
    for (int n = 0; n < NBOX; ++n) {
      if (cBox[n] == tid) { ws += pBox[n]; cnt += vBox[n]; }
    }
    float val;
    if (cnt > 0.0f) {
      val = ws / (cnt < 1.0f ? 1.0f : cnt);
    } else {
      const int ny = negy[b * NUM_CLASSES + tid];
      const int nx = negx[b * NUM_CLASSES + tid];
      val = plane[ny * WB + nx];
    }
    out[(size_t)(b * NUM_CLASSES + tid) * CCH + c] = val;
  }
}

extern "C" void kernel_launch(void* const* d_in, const int* in_sizes, int n_in,
                              void* d_out, int out_size, void* d_ws, size_t ws_size,
                              hipStream_t stream) {
  (void)in_sizes; (void)n_in; (void)out_size; (void)d_ws; (void)ws_size;
  const float* feat  = (const float*)d_in[0];
  const float* boxes = (const float*)d_in[1];
  const int*   gtc   = (const int*)d_in[2];
  const int*   negy  = (const int*)d_in[3];
  const int*   negx  = (const int*)d_in[4];
  const int*   imh   = (const int*)d_in[5];
  const int*   imw   = (const int*)d_in[6];
  float* out = (float*)d_out;
  const int nblocks = 8 * CCH;   // one workgroup per (image, channel) plane
  vpe_sat_wmma_kernel<<<nblocks, 256, 0, stream>>>(feat, boxes, gtc, negy, negx, imh, imw, out);
}